// RISDA_CE_3143916060778
// MI455X (gfx1250) — compile-verified
//
#include <hip/hip_runtime.h>
#include <hip/hip_bf16.h>

#define NN    1024
#define CC    1000
#define AA    64
#define HEADN 100
#define MTAIL (CC - HEADN)   // 900

typedef float v2f __attribute__((ext_vector_type(2)));
typedef float v8f __attribute__((ext_vector_type(8)));

__device__ __forceinline__ v8f wmma_f32_16x16x4(v2f a, v2f b, v8f c) {
  // 8 args: (neg_a, A, neg_b, B, c_mod, C, reuse_a, reuse_b)
  return __builtin_amdgcn_wmma_f32_16x16x4_f32(false, a, false, b, (short)0, c,
                                               false, false);
}

// ---------------------------------------------------------------------------
// K1: per-class statistics -> cov_upd[c] (EstimatorCV.update_CV, forward only)
// ---------------------------------------------------------------------------
__global__ void k_stats_cov(const float* __restrict__ feat,
                            const int*   __restrict__ labels,
                            const float* __restrict__ cov_state,
                            const float* __restrict__ ave_state,
                            const float* __restrict__ amount_state,
                            float* __restrict__ cov_upd) {
  const int c   = blockIdx.x;
  const int tid = threadIdx.x;            // 256 threads
  __shared__ int   sl[NN];
  __shared__ float sa[AA];                // sum -> ave
  __shared__ float sd[AA];                // centered row
  __shared__ float sda[AA];               // ave_state - ave
  __shared__ float scnt, s_wcv, s_cnts;

  for (int n = tid; n < NN; n += 256) sl[n] = labels[n];
  if (tid < AA) sa[tid] = 0.f;
  if (tid == 0) scnt = 0.f;
  __syncthreads();

  for (int n = tid; n < NN; n += 256) {
    if (sl[n] == c) {
      atomicAdd(&scnt, 1.0f);
      for (int a = 0; a < AA; ++a) atomicAdd(&sa[a], feat[n * AA + a]);
    }
  }
  __syncthreads();

  if (tid == 0) {
    float cnt   = scnt;
    float cnts  = (cnt == 0.f) ? 1.f : cnt;
    float denom = cnt + amount_state[c];
    s_wcv  = (denom > 0.f) ? (cnt / denom) : 0.f;
    s_cnts = cnts;
  }
  __syncthreads();
  if (tid < AA) {
    float av = sa[tid] / s_cnts;
    sa[tid]  = av;
    sda[tid] = ave_state[c * AA + tid] - av;
  }
  __syncthreads();

  float acc[16];
#pragma unroll
  for (int j = 0; j < 16; ++j) acc[j] = 0.f;

  for (int n = 0; n < NN; ++n) {
    if (sl[n] == c) {                      // uniform across block
      if (tid < AA) sd[tid] = feat[n * AA + tid] - sa[tid];
      __syncthreads();
#pragma unroll
      for (int j = 0; j < 16; ++j) {
        int idx = tid + j * 256;
        acc[j] += sd[idx >> 6] * sd[idx & 63];
      }
      __syncthreads();
    }
  }

  const float wcv = s_wcv, cnts = s_cnts;
#pragma unroll
  for (int j = 0; j < 16; ++j) {
    int idx = tid + j * 256;
    int a = idx >> 6, b = idx & 63;
    float v = cov_state[(size_t)c * AA * AA + idx] * (1.f - wcv)
            + (acc[j] / cnts) * wcv
            + wcv * (1.f - wcv) * sda[a] * sda[b];
    cov_upd[(size_t)c * AA * AA + idx] = v;
  }
}

// ---------------------------------------------------------------------------
// K2: tail_cov = kg[head:, :] @ cov_upd_flat   (900 x 1000) x (1000 x 4096)
// fp32 WMMA 16x16x4. Block = 8 waves = 32(M) x 256(N) tile.
// Each wave: one A fragment reused across 4 accumulators (16 x 64 per wave).
// ---------------------------------------------------------------------------
__global__ void k_kg_gemm(const float* __restrict__ kg,
                          const float* __restrict__ cov_upd,
                          float* __restrict__ tailcov) {
  const int tid  = threadIdx.x;          // 256
  const int wave = tid >> 5, lane = tid & 31;
  const int half = lane >> 4, l16 = lane & 15;
  const int mw = wave >> 2, nw = wave & 3;
  const int m0 = blockIdx.x * 32 + mw * 16;
  const int n0 = blockIdx.y * 256 + nw * 64;

  const float* kgtail = kg + (size_t)HEADN * CC;
  int ra = m0 + l16; if (ra > MTAIL - 1) ra = MTAIL - 1;  // clamp, store-guarded
  const float* Aptr = kgtail + (size_t)ra * CC;

  v8f acc0 = {0.f,0.f,0.f,0.f,0.f,0.f,0.f,0.f};
  v8f acc1 = acc0, acc2 = acc0, acc3 = acc0;

  for (int k0 = 0; k0 < CC; k0 += 4) {
    const int ka = k0 + 2 * half;
    v2f a;
    a.x = Aptr[ka];
    a.y = Aptr[ka + 1];
    const float* B0 = cov_upd + (size_t)ka * (AA * AA) + n0 + l16;
    const float* B1 = cov_upd + (size_t)(ka + 1) * (AA * AA) + n0 + l16;
    v2f b;
    b.x = B0[0];  b.y = B1[0];  acc0 = wmma_f32_16x16x4(a, b, acc0);
    b.x = B0[16]; b.y = B1[16]; acc1 = wmma_f32_16x16x4(a, b, acc1);
    b.x = B0[32]; b.y = B1[32]; acc2 = wmma_f32_16x16x4(a, b, acc2);
    b.x = B0[48]; b.y = B1[48]; acc3 = wmma_f32_16x16x4(a, b, acc3);
  }
#pragma unroll
  for (int r = 0; r < 8; ++r) {
    const int row = m0 + r + 8 * half;
    if (row < MTAIL) {
      const size_t base = (size_t)row * (AA * AA) + n0 + l16;
      tailcov[base]      = acc0[r];
      tailcov[base + 16] = acc1[r];
      tailcov[base + 32] = acc2[r];
      tailcov[base + 48] = acc3[r];
    }
  }
}

// ---------------------------------------------------------------------------
// K3: per-class scalars q_l = w_l^T S_l w_l, s_l = w_l . o_l
// ---------------------------------------------------------------------------
__global__ void k_qs(const float* __restrict__ W,
                     const float* __restrict__ out_new,
                     const float* __restrict__ cov_upd,
                     const float* __restrict__ tailcov,
                     float* __restrict__ qs, float* __restrict__ ss) {
  const int l = blockIdx.x;
  const int a = threadIdx.x;   // 64
  const float* S = (l < HEADN) ? (cov_upd + (size_t)l * AA * AA)
                               : (tailcov + (size_t)(l - HEADN) * AA * AA);
  __shared__ float wl[AA], rq[AA], rs[AA];
  wl[a] = W[l * AA + a];
  __syncthreads();
  float u = 0.f;
  for (int b = 0; b < AA; ++b) u += S[a * AA + b] * wl[b];
  rq[a] = wl[a] * u;
  rs[a] = wl[a] * out_new[l * AA + a];
  __syncthreads();
  for (int off = 32; off > 0; off >>= 1) {
    if (a < off) { rq[a] += rq[a + off]; rs[a] += rs[a + off]; }
    __syncthreads();
  }
  if (a == 0) { qs[l] = rq[0]; ss[l] = rs[0]; }
}

// ---------------------------------------------------------------------------
// K4: row[l,c] = 0.5*beta*(w_c^T S_l w_c - 2 w_c^T S_l w_l + q_l)
//              + alpha*(w_c.o_l - s_l)
// Y = W @ S_l via WMMA. Block = 4 waves, each wave: 16 classes x full N=64
// (4 accumulators, shared A fragment). 64 classes per block.
// ---------------------------------------------------------------------------
__global__ void k_rowfull(const float* __restrict__ W,
                          const float* __restrict__ out_new,
                          const float* __restrict__ cov_upd,
                          const float* __restrict__ tailcov,
                          const float* __restrict__ qs,
                          const float* __restrict__ ss,
                          const float* __restrict__ alphap,
                          const float* __restrict__ betap,
                          float* __restrict__ rowfull) {
  const int l   = blockIdx.y;
  const int tid = threadIdx.x;            // 128
  const int wave = tid >> 5, lane = tid & 31;
  const int half = lane >> 4, l16 = lane & 15;
  const int m0 = blockIdx.x * 64 + wave * 16;   // class tile of this wave
  const float* S = (l < HEADN) ? (cov_upd + (size_t)l * AA * AA)
                               : (tailcov + (size_t)(l - HEADN) * AA * AA);
  __shared__ float Y[64][AA + 1];
  __shared__ float wl[AA], ol[AA];
  __shared__ float rd[64][2], rg[64][2], rm[64][2];
  if (tid < AA) { wl[tid] = W[l * AA + tid]; ol[tid] = out_new[l * AA + tid]; }

  int ra = m0 + l16; if (ra > CC - 1) ra = CC - 1;
  const float* Aptr = W + (size_t)ra * AA;

  v8f acc0 = {0.f,0.f,0.f,0.f,0.f,0.f,0.f,0.f};
  v8f acc1 = acc0, acc2 = acc0, acc3 = acc0;
  for (int k0 = 0; k0 < AA; k0 += 4) {
    const int ka = k0 + 2 * half;
    v2f a;
    a.x = Aptr[ka];
    a.y = Aptr[ka + 1];
    const float* B0 = S + ka * AA + l16;
    const float* B1 = S + (ka + 1) * AA + l16;
    v2f b;
    b.x = B0[0];  b.y = B1[0];  acc0 = wmma_f32_16x16x4(a, b, acc0);
    b.x = B0[16]; b.y = B1[16]; acc1 = wmma_f32_16x16x4(a, b, acc1);
    b.x = B0[32]; b.y = B1[32]; acc2 = wmma_f32_16x16x4(a, b, acc2);
    b.x = B0[48]; b.y = B1[48]; acc3 = wmma_f32_16x16x4(a, b, acc3);
  }
  const int lrow = wave * 16;
#pragma unroll
  for (int r = 0; r < 8; ++r) {
    const int rr = lrow + r + 8 * half;
    Y[rr][l16]      = acc0[r];
    Y[rr][l16 + 16] = acc1[r];
    Y[rr][l16 + 32] = acc2[r];
    Y[rr][l16 + 48] = acc3[r];
  }
  __syncthreads();

  // d/g/m reductions: 2 lanes per class row, 32 elements each
  const int row = tid >> 1;          // 0..63
  const int sub = tid & 1;
  const int c   = blockIdx.x * 64 + row;
  const int csafe = (c < CC) ? c : (CC - 1);
  const float* wc = W + (size_t)csafe * AA;
  float d = 0.f, g = 0.f, m = 0.f;
  for (int a2 = sub; a2 < AA; a2 += 2) {
    float ya = Y[row][a2];
    float w2 = wc[a2];
    d += ya * w2;
    g += ya * wl[a2];
    m += w2 * ol[a2];
  }
  rd[row][sub] = d; rg[row][sub] = g; rm[row][sub] = m;
  __syncthreads();
  if (sub == 0 && c < CC) {
    const float beta = *betap, alpha = *alphap;
    float dt = rd[row][0] + rd[row][1];
    float gt = rg[row][0] + rg[row][1];
    float mt = rm[row][0] + rm[row][1];
    rowfull[(size_t)l * CC + c] =
        0.5f * beta * (dt - 2.f * gt + qs[l]) + alpha * (mt - ss[l]);
  }
}

// ---------------------------------------------------------------------------
// K5: aug = y_s[n,:] + row[labels[n],:]; weighted CE accumulate (two atomics)
// ---------------------------------------------------------------------------
__global__ void k_ce(const float* __restrict__ y_s,
                     const float* __restrict__ rowfull,
                     const int*   __restrict__ labels,
                     const float* __restrict__ weights,
                     float* __restrict__ numden) {
  const int n   = blockIdx.x;
  const int tid = threadIdx.x;    // 256
  __shared__ float aug[CC];
  __shared__ float red[256];
  const int l = labels[n];

  float lmax = -3.4e38f;
  for (int c = tid; c < CC; c += 256) {
    float v = y_s[(size_t)n * CC + c] + rowfull[(size_t)l * CC + c];
    aug[c] = v;
    lmax = fmaxf(lmax, v);
  }
  red[tid] = lmax; __syncthreads();
  for (int off = 128; off > 0; off >>= 1) {
    if (tid < off) red[tid] = fmaxf(red[tid], red[tid + off]);
    __syncthreads();
  }
  const float m = red[0];
  __syncthreads();

  float ls = 0.f;
  for (int c = tid; c < CC; c += 256) ls += __expf(aug[c] - m);
  red[tid] = ls; __syncthreads();
  for (int off = 128; off > 0; off >>= 1) {
    if (tid < off) red[tid] += red[tid + off];
    __syncthreads();
  }
  if (tid == 0) {
    float logp = aug[l] - m - logf(red[0]);
    float w = weights[l];
    atomicAdd(&numden[0], w * (-logp));
    atomicAdd(&numden[1], w);
  }
}

__global__ void k_final(const float* __restrict__ numden, float* __restrict__ out) {
  out[0] = numden[0] / numden[1];
}

// ---------------------------------------------------------------------------
extern "C" void kernel_launch(void* const* d_in, const int* in_sizes, int n_in,
                              void* d_out, int out_size, void* d_ws, size_t ws_size,
                              hipStream_t stream) {
  const float* feat         = (const float*)d_in[0];   // [N,A]
  const float* y_s          = (const float*)d_in[1];   // [N,C]
  const float* weights      = (const float*)d_in[2];   // [C]
  const float* kg           = (const float*)d_in[3];   // [C,C]
  const float* out_new      = (const float*)d_in[4];   // [C,A]
  const float* fc_w         = (const float*)d_in[5];   // [C,A]
  const float* alpha        = (const float*)d_in[6];   // [1]
  const float* beta         = (const float*)d_in[7];   // [1]
  const float* cov_state    = (const float*)d_in[8];   // [C,A,A]
  const float* ave_state    = (const float*)d_in[9];   // [C,A]
  const float* amount_state = (const float*)d_in[10];  // [C]
  const int*   labels       = (const int*)d_in[11];    // [N]
  // d_in[12] = head (compile-time HEADN = 100, fixed problem shape)

  float* ws      = (float*)d_ws;
  float* cov_upd = ws;                                   // C*A*A
  float* tailcov = cov_upd + (size_t)CC * AA * AA;       // (C-head)*A*A
  float* rowfull = tailcov + (size_t)MTAIL * AA * AA;    // C*C
  float* qs      = rowfull + (size_t)CC * CC;            // C
  float* ss      = qs + CC;                              // C
  float* numden  = ss + CC;                              // 2

  hipMemsetAsync(numden, 0, 2 * sizeof(float), stream);

  k_stats_cov<<<CC, 256, 0, stream>>>(feat, labels, cov_state, ave_state,
                                      amount_state, cov_upd);
  k_kg_gemm<<<dim3((MTAIL + 31) / 32, (AA * AA) / 256), 256, 0, stream>>>(
      kg, cov_upd, tailcov);
  k_qs<<<CC, 64, 0, stream>>>(fc_w, out_new, cov_upd, tailcov, qs, ss);
  k_rowfull<<<dim3((CC + 63) / 64, CC), 128, 0, stream>>>(
      fc_w, out_new, cov_upd, tailcov, qs, ss, alpha, beta, rowfull);
  k_ce<<<NN, 256, 0, stream>>>(y_s, rowfull, labels, weights, numden);
  k_final<<<1, 1, 0, stream>>>(numden, (float*)d_out);
}